// FFNTM_78477642433000
// MI455X (gfx1250) — compile-verified
//
#include <hip/hip_runtime.h>
#include <math.h>

typedef float v2f __attribute__((ext_vector_type(2)));
typedef float v8f __attribute__((ext_vector_type(8)));

#define NLOC 262144
#define DIM  128
#define CDIM 512
#define NHEADS 3   // 2 read + 1 write
#define EPS 1e-8f

// params region offsets (floats) inside workspace
#define P_C     0      // 512
#define P_K     512    // 3*128
#define P_KN    896    // 3
#define P_BETA  900    // 3
#define P_G     904    // 3
#define P_S     908    // 9
#define P_GAMMA 920    // 3
#define P_MAXZ  924    // 3 (ordered uint)
#define P_SUMZ  928    // 3
#define P_MAXR  932    // 3 (ordered uint)
#define P_SUMR  936    // 3
#define P_SUMWP 940    // 3
#define P_E     944    // 128
#define P_A     1072   // 128
#define P_SIZE  2048

// d_out layout (floats): out[256] | readings[256] | read_w[2N] | write_w[N] | M_new[N*128]
#define O_READINGS 256
#define O_READW    512
#define O_WRITEW   (512 + 2*NLOC)
#define O_MNEW     (512 + 3*NLOC)

__device__ __forceinline__ unsigned f2ord(float f) {
    unsigned u = __float_as_uint(f);
    return (u & 0x80000000u) ? ~u : (u | 0x80000000u);
}
__device__ __forceinline__ float ord2f(unsigned e) {
    unsigned u = (e & 0x80000000u) ? (e & 0x7fffffffu) : ~e;
    return __uint_as_float(u);
}
__device__ __forceinline__ float softplusf(float x) {
    return fmaxf(x, 0.f) + log1pf(expf(-fabsf(x)));
}
__device__ __forceinline__ float sigmoidf(float x) { return 1.f / (1.f + expf(-x)); }
__device__ __forceinline__ float dot512(const float* __restrict__ w, const float* __restrict__ c) {
    float s = 0.f;
    #pragma unroll 8
    for (int j = 0; j < CDIM; ++j) s += w[j] * c[j];
    return s;
}

// ---------------- tiny setup kernels ----------------

__global__ void k_init(float* Pp) {
    int t = threadIdx.x;
    if (t < NHEADS) {
        ((unsigned*)Pp)[P_MAXZ + t] = 0u;   // ordered encoding of -inf-ish
        ((unsigned*)Pp)[P_MAXR + t] = 0u;
    }
}

__global__ void k_controller(const float* __restrict__ ein, const float* __restrict__ pr,
                             const float* __restrict__ Wc, const float* __restrict__ bc,
                             float* __restrict__ Pp) {
    int i = blockIdx.x * blockDim.x + threadIdx.x;
    if (i < CDIM) {
        const float* row = Wc + (size_t)i * 512;
        float s = bc[i];
        for (int j = 0; j < 256; ++j) s += row[j] * ein[j];
        for (int j = 0; j < 256; ++j) s += row[256 + j] * pr[j];
        Pp[P_C + i] = s;
    }
}

__global__ void k_heads(const float* rWk, const float* rbk, const float* rWb, const float* rbb,
                        const float* rWg, const float* rbg, const float* rWs, const float* rbs,
                        const float* rWgm, const float* rbgm,
                        const float* wWk, const float* wbk, const float* wWb, const float* wbb,
                        const float* wWg, const float* wbg, const float* wWs, const float* wbs,
                        const float* wWgm, const float* wbgm,
                        const float* wWe, const float* wbe, const float* wWa, const float* wba,
                        float* Pp) {
    int t = blockIdx.x * blockDim.x + threadIdx.x;
    const float* c = Pp + P_C;
    if (t < 384) {                                 // keys
        int h = t >> 7, d = t & 127;
        const float* Wrow; float b;
        if (h < 2) { Wrow = rWk + ((size_t)h * DIM + d) * CDIM; b = rbk[h * DIM + d]; }
        else       { Wrow = wWk + (size_t)d * CDIM;             b = wbk[d]; }
        Pp[P_K + h * DIM + d] = tanhf(dot512(Wrow, c) + b);
    } else if (t < 387) {                          // beta
        int h = t - 384;
        const float* Wrow = (h < 2) ? rWb + (size_t)h * CDIM : wWb;
        float b = (h < 2) ? rbb[h] : wbb[0];
        Pp[P_BETA + h] = softplusf(dot512(Wrow, c) + b);
    } else if (t < 390) {                          // gate g
        int h = t - 387;
        const float* Wrow = (h < 2) ? rWg + (size_t)h * CDIM : wWg;
        float b = (h < 2) ? rbg[h] : wbg[0];
        Pp[P_G + h] = sigmoidf(dot512(Wrow, c) + b);
    } else if (t < 393) {                          // shift softmax (S=3)
        int h = t - 390;
        float l[3];
        for (int j = 0; j < 3; ++j) {
            const float* Wrow = (h < 2) ? rWs + ((size_t)h * 3 + j) * CDIM : wWs + (size_t)j * CDIM;
            float b = (h < 2) ? rbs[h * 3 + j] : wbs[j];
            l[j] = dot512(Wrow, c) + b;
        }
        float m = fmaxf(l[0], fmaxf(l[1], l[2]));
        float e0 = expf(l[0] - m), e1 = expf(l[1] - m), e2 = expf(l[2] - m);
        float s = e0 + e1 + e2;
        Pp[P_S + h * 3 + 0] = e0 / s;
        Pp[P_S + h * 3 + 1] = e1 / s;
        Pp[P_S + h * 3 + 2] = e2 / s;
    } else if (t < 396) {                          // gamma
        int h = t - 393;
        const float* Wrow = (h < 2) ? rWgm + (size_t)h * CDIM : wWgm;
        float b = (h < 2) ? rbgm[h] : wbgm[0];
        Pp[P_GAMMA + h] = 1.f + softplusf(dot512(Wrow, c) + b);
    } else if (t >= 512 && t < 640) {              // erase vector
        int d = t - 512;
        Pp[P_E + d] = sigmoidf(dot512(wWe + (size_t)d * CDIM, c) + wbe[d]);
    } else if (t >= 640 && t < 768) {              // add vector
        int d = t - 640;
        Pp[P_A + d] = tanhf(dot512(wWa + (size_t)d * CDIM, c) + wba[d]);
    }
}

__global__ void k_knorm(float* Pp) {
    __shared__ float red[128];
    int t = threadIdx.x;
    for (int h = 0; h < NHEADS; ++h) {
        float v = Pp[P_K + h * DIM + t];
        red[t] = v * v;
        __syncthreads();
        for (int s = 64; s > 0; s >>= 1) { if (t < s) red[t] += red[t + s]; __syncthreads(); }
        if (t == 0) Pp[P_KN + h] = sqrtf(red[0]);
        __syncthreads();
    }
}

// ---------------- big pass 1: WMMA content addressing ----------------
// A = M rows (16x4 f32), B = keys padded to 16 cols (3 live), C = sim[16 rows][16 heads].
__global__ void k_simpass(const float* __restrict__ M, const float* __restrict__ rinit,
                          const float* __restrict__ winit, float* __restrict__ Pp,
                          float* __restrict__ buf0) {
    __shared__ float sk[16 * DIM];                 // zero-padded: heads 3..15 are zero rows
    int tid = threadIdx.x;
    for (int i = tid; i < 16 * DIM; i += blockDim.x)
        sk[i] = (i < NHEADS * DIM) ? Pp[P_K + i] : 0.f;
    __syncthreads();

    int lane = tid & 31;
    int wave = blockIdx.x * (blockDim.x >> 5) + (tid >> 5);
    int h = lane & 15;
    int half = lane >> 4;
    float beta = (h < NHEADS) ? Pp[P_BETA + h] : 0.f;
    float kn   = (h < NHEADS) ? Pp[P_KN + h]   : 1.f;
    float zmax = -3.4e38f, rmax = -3.4e38f;

    const float* skrow = sk + h * DIM + 2 * half;  // unconditional, always in-bounds
    int rowstart = wave * 256;                     // 1024 waves x 256 rows = N
    for (int gbase = rowstart; gbase < rowstart + 256; gbase += 16) {
        v8f acc = {};
        float sq = 0.f;
        const float* arow = M + (size_t)(gbase + (lane & 15)) * DIM + 2 * half;
        #pragma unroll 4
        for (int d0 = 0; d0 < DIM; d0 += 4) {
            float2 mv = *(const float2*)(arow + d0);
            float2 kv = *(const float2*)(skrow + d0);
            sq += mv.x * mv.x + mv.y * mv.y;
            v2f A; A.x = mv.x; A.y = mv.y;
            v2f B; B.x = kv.x; B.y = kv.y;
            acc = __builtin_amdgcn_wmma_f32_16x16x4_f32(false, A, false, B, (short)0, acc,
                                                        false, false);
        }
        float rowsq = sq + __shfl_xor(sq, 16, 32);  // lane L holds ||row (L&15)||^2
        #pragma unroll
        for (int r = 0; r < 8; ++r) {
            int rowoff = r + 8 * half;
            float msq = __shfl(rowsq, rowoff, 32);
            if (h < NHEADS) {
                int n = gbase + rowoff;
                float z = beta * acc[r] / (sqrtf(msq) * kn + EPS);
                buf0[h * NLOC + n] = z;
                zmax = fmaxf(zmax, z);
                float lr = (h < 2) ? rinit[h * NLOC + n] : winit[n];
                rmax = fmaxf(rmax, lr);
            }
        }
    }
    zmax = fmaxf(zmax, __shfl_xor(zmax, 16, 32));
    rmax = fmaxf(rmax, __shfl_xor(rmax, 16, 32));
    if (h < NHEADS && half == 0) {
        atomicMax((unsigned*)Pp + P_MAXZ + h, f2ord(zmax));
        atomicMax((unsigned*)Pp + P_MAXR + h, f2ord(rmax));
    }
}

// ---------------- softmax sums (deterministic block trees) ----------------
__global__ void k_expsum(const float* __restrict__ buf0, const float* __restrict__ rinit,
                         const float* __restrict__ winit, const float* __restrict__ Pp,
                         float* __restrict__ expart) {
    __shared__ float sdata[256];
    int tid = threadIdx.x;
    int n = blockIdx.x * blockDim.x + tid;
    float loc[6];
    for (int h = 0; h < NHEADS; ++h) {
        float mz = ord2f(((const unsigned*)Pp)[P_MAXZ + h]);
        float mr = ord2f(((const unsigned*)Pp)[P_MAXR + h]);
        loc[h]     = expf(buf0[h * NLOC + n] - mz);
        float lr   = (h < 2) ? rinit[h * NLOC + n] : winit[n];
        loc[3 + h] = expf(lr - mr);
    }
    for (int q = 0; q < 6; ++q) {
        sdata[tid] = loc[q];
        __syncthreads();
        for (int s = 128; s > 0; s >>= 1) { if (tid < s) sdata[tid] += sdata[tid + s]; __syncthreads(); }
        if (tid == 0) expart[q * 1024 + blockIdx.x] = sdata[0];
        __syncthreads();
    }
}

__global__ void k_fin1(const float* __restrict__ expart, float* __restrict__ Pp) {
    int t = threadIdx.x;
    if (t < 6) {
        float s = 0.f;
        for (int i = 0; i < 1024; ++i) s += expart[t * 1024 + i];
        if (t < 3) Pp[P_SUMZ + t] = s; else Pp[P_SUMR + t - 3] = s;
    }
}

__global__ void k_wg(float* __restrict__ buf0, const float* __restrict__ rinit,
                     const float* __restrict__ winit, const float* __restrict__ Pp) {
    int n = blockIdx.x * blockDim.x + threadIdx.x;
    for (int h = 0; h < NHEADS; ++h) {
        float mz = ord2f(((const unsigned*)Pp)[P_MAXZ + h]);
        float mr = ord2f(((const unsigned*)Pp)[P_MAXR + h]);
        float wc = expf(buf0[h * NLOC + n] - mz) / Pp[P_SUMZ + h];
        float lr = (h < 2) ? rinit[h * NLOC + n] : winit[n];
        float wp = expf(lr - mr) / Pp[P_SUMR + h];
        float g = Pp[P_G + h];
        buf0[h * NLOC + n] = g * wc + (1.f - g) * wp;
    }
}

__global__ void k_shift(const float* __restrict__ buf0, float* __restrict__ buf1,
                        const float* __restrict__ Pp, float* __restrict__ wppart) {
    __shared__ float sdata[256];
    int tid = threadIdx.x;
    int n = blockIdx.x * blockDim.x + tid;
    int np1 = (n + 1) & (NLOC - 1);
    int nm1 = (n - 1) & (NLOC - 1);
    float loc[3];
    for (int h = 0; h < NHEADS; ++h) {
        int base = h * NLOC;
        float wt = Pp[P_S + h * 3 + 0] * buf0[base + np1]
                 + Pp[P_S + h * 3 + 1] * buf0[base + n]
                 + Pp[P_S + h * 3 + 2] * buf0[base + nm1];
        float wp = powf(wt, Pp[P_GAMMA + h]);
        buf1[base + n] = wp;
        loc[h] = wp;
    }
    for (int q = 0; q < 3; ++q) {
        sdata[tid] = loc[q];
        __syncthreads();
        for (int s = 128; s > 0; s >>= 1) { if (tid < s) sdata[tid] += sdata[tid + s]; __syncthreads(); }
        if (tid == 0) wppart[q * 1024 + blockIdx.x] = sdata[0];
        __syncthreads();
    }
}

__global__ void k_fin2(const float* __restrict__ wppart, float* __restrict__ Pp) {
    int t = threadIdx.x;
    if (t < 3) {
        float s = 0.f;
        for (int i = 0; i < 1024; ++i) s += wppart[t * 1024 + i];
        Pp[P_SUMWP + t] = s;
    }
}

__global__ void k_norm(const float* __restrict__ buf1, const float* __restrict__ Pp,
                       float* __restrict__ out) {
    int n = blockIdx.x * blockDim.x + threadIdx.x;
    for (int h = 0; h < NHEADS; ++h) {
        float w = buf1[h * NLOC + n] / (Pp[P_SUMWP + h] + EPS);
        if (h < 2) out[O_READW + h * NLOC + n] = w;
        else       out[O_WRITEW + n] = w;
    }
}

// ---------------- big pass 2: WMMA readings + fused M_new write ----------------
// A = read weights (16x4, rows 0..1 live, masked), B = M (4 rows x 16 D-cols).
__global__ void k_pass2(const float* __restrict__ M, const float* __restrict__ Pp,
                        float* __restrict__ out, float* __restrict__ rdpart) {
    const float* rw  = out + O_READW;
    const float* wwv = out + O_WRITEW;
    float* mnew      = out + O_MNEW;

    int tid = threadIdx.x, lane = tid & 31;
    int W = blockIdx.x * (blockDim.x >> 5) + (tid >> 5);   // 1024 waves
    int dt = W & 7;            // D tile (8 x 16 cols)
    int chunk = W >> 3;        // 128 chunks x 2048 rows
    int half = lane >> 4;
    int head = lane & 15;
    int d = dt * 16 + head;
    float e_l = Pp[P_E + d], a_l = Pp[P_A + d];
    int hsel = (head < 2) ? head : 1;              // clamped, always in-bounds
    float amask = (head < 2) ? 1.f : 0.f;
    const float* rwrow = rw + (size_t)hsel * NLOC;

    v8f acc = {};
    int nbase = chunk * 2048;
    #pragma unroll 4
    for (int n0 = nbase; n0 < nbase + 2048; n0 += 4) {
        int nb = n0 + 2 * half;
        float b0 = M[(size_t)nb * DIM + d];
        float b1 = M[(size_t)(nb + 1) * DIM + d];
        float2 av = *(const float2*)(rwrow + nb);
        float2 wv = *(const float2*)(wwv + nb);
        v2f A, B;
        A.x = av.x * amask; A.y = av.y * amask;
        B.x = b0; B.y = b1;
        acc = __builtin_amdgcn_wmma_f32_16x16x4_f32(false, A, false, B, (short)0, acc,
                                                    false, false);
        mnew[(size_t)nb * DIM + d]       = b0 * (1.f - wv.x * e_l) + wv.x * a_l;
        mnew[(size_t)(nb + 1) * DIM + d] = b1 * (1.f - wv.y * e_l) + wv.y * a_l;
    }
    if (lane < 16) {
        rdpart[chunk * 256 + dt * 16 + lane]       = acc[0];   // head 0
        rdpart[chunk * 256 + 128 + dt * 16 + lane] = acc[1];   // head 1
    }
}

__global__ void k_reduce_readings(const float* __restrict__ rdpart, float* __restrict__ out) {
    int t = threadIdx.x;   // t = h*128 + d
    float s = 0.f;
    for (int ch = 0; ch < 128; ++ch) s += rdpart[ch * 256 + t];
    out[O_READINGS + t] = s;
}

__global__ void k_out(const float* __restrict__ Wo, const float* __restrict__ bo,
                      const float* __restrict__ Pp, float* __restrict__ out) {
    int i = threadIdx.x;
    const float* rd = out + O_READINGS;
    const float* row = Wo + (size_t)i * 768;
    float s = bo[i];
    for (int j = 0; j < 512; ++j) s += row[j] * Pp[P_C + j];
    for (int j = 0; j < 256; ++j) s += row[512 + j] * rd[j];
    out[i] = s;
}

extern "C" void kernel_launch(void* const* d_in, const int* in_sizes, int n_in,
                              void* d_out, int out_size, void* d_ws, size_t ws_size,
                              hipStream_t stream) {
    (void)in_sizes; (void)n_in; (void)out_size; (void)ws_size;
    const float* ein   = (const float*)d_in[0];
    const float* M     = (const float*)d_in[1];
    const float* pr    = (const float*)d_in[2];
    const float* Wc    = (const float*)d_in[3];
    const float* bc    = (const float*)d_in[4];
    const float* Wo    = (const float*)d_in[5];
    const float* bo    = (const float*)d_in[6];
    const float* rWk   = (const float*)d_in[7];
    const float* rbk   = (const float*)d_in[8];
    const float* rWb   = (const float*)d_in[9];
    const float* rbb   = (const float*)d_in[10];
    const float* rWg   = (const float*)d_in[11];
    const float* rbg   = (const float*)d_in[12];
    const float* rWs   = (const float*)d_in[13];
    const float* rbs   = (const float*)d_in[14];
    const float* rWgm  = (const float*)d_in[15];
    const float* rbgm  = (const float*)d_in[16];
    const float* rinit = (const float*)d_in[17];
    const float* wWk   = (const float*)d_in[18];
    const float* wbk   = (const float*)d_in[19];
    const float* wWb   = (const float*)d_in[20];
    const float* wbb   = (const float*)d_in[21];
    const float* wWg   = (const float*)d_in[22];
    const float* wbg   = (const float*)d_in[23];
    const float* wWs   = (const float*)d_in[24];
    const float* wbs   = (const float*)d_in[25];
    const float* wWgm  = (const float*)d_in[26];
    const float* wbgm  = (const float*)d_in[27];
    const float* wWe   = (const float*)d_in[28];
    const float* wbe   = (const float*)d_in[29];
    const float* wWa   = (const float*)d_in[30];
    const float* wba   = (const float*)d_in[31];
    const float* winit = (const float*)d_in[32];

    float* out = (float*)d_out;
    float* wsf = (float*)d_ws;
    float* buf0   = wsf;
    float* buf1   = wsf + 3 * NLOC;
    float* Pp     = wsf + 6 * NLOC;
    float* expart = Pp + P_SIZE;          // 6*1024
    float* wppart = expart + 6 * 1024;    // 3*1024
    float* rdpart = wppart + 3 * 1024;    // 128*256

    k_init<<<1, 64, 0, stream>>>(Pp);
    k_controller<<<2, 256, 0, stream>>>(ein, pr, Wc, bc, Pp);
    k_heads<<<4, 256, 0, stream>>>(rWk, rbk, rWb, rbb, rWg, rbg, rWs, rbs, rWgm, rbgm,
                                   wWk, wbk, wWb, wbb, wWg, wbg, wWs, wbs, wWgm, wbgm,
                                   wWe, wbe, wWa, wba, Pp);
    k_knorm<<<1, 128, 0, stream>>>(Pp);
    k_simpass<<<128, 256, 0, stream>>>(M, rinit, winit, Pp, buf0);
    k_expsum<<<1024, 256, 0, stream>>>(buf0, rinit, winit, Pp, expart);
    k_fin1<<<1, 64, 0, stream>>>(expart, Pp);
    k_wg<<<1024, 256, 0, stream>>>(buf0, rinit, winit, Pp);
    k_shift<<<1024, 256, 0, stream>>>(buf0, buf1, Pp, wppart);
    k_fin2<<<1, 64, 0, stream>>>(wppart, Pp);
    k_norm<<<1024, 256, 0, stream>>>(buf1, Pp, out);
    k_pass2<<<128, 256, 0, stream>>>(M, Pp, out, rdpart);
    k_reduce_readings<<<1, 256, 0, stream>>>(rdpart, out);
    k_out<<<1, 256, 0, stream>>>(Wo, bo, Pp, out);
}